// WaveletTransform3D_68212670595876
// MI455X (gfx1250) — compile-verified
//
#include <hip/hip_runtime.h>
#include <hip/hip_bf16.h>
#include <cstddef>

// ---------------------------------------------------------------------------
// 3D Haar wavedec, 5 levels, on [4,8,128,128,128] fp32.
// Each level: one WMMA tile kernel (16^3 tiles, 1 wave32 per tile) doing the
// separable transform as three batched 16x16 GEMMs against the constant Haar
// matrix via V_WMMA_F32_16X16X4_F32. Level 5 (8^3 -> 4^3) is a tiny scalar
// kernel. aaa bands ping-pong through d_ws.
// ---------------------------------------------------------------------------

typedef float v2f __attribute__((ext_vector_type(2)));
typedef float v8f __attribute__((ext_vector_type(8)));

struct DetPtrs { float* p[7]; };

#define HAAR_C 0.70710678118654752440f
// LDS tile layout: [16][16][16] padded -> strides (320, 20, 1) floats.
// 20 mod 64: gcd(20,64)=4 -> 16 distinct banks across 16 lanes; w4 offsets
// stay 16B-aligned for float4 LDS ops.
#define SD 320
#define SH 20

// Haar analysis matrix H[n][k] (16x16): rows 0-7 = lo, rows 8-15 = hi.
__device__ __forceinline__ float haarB(int n, int k) {
  int base = (n & 7) * 2;
  if (k == base)     return HAAR_C;
  if (k == base + 1) return (n < 8) ? HAAR_C : -HAAR_C;
  return 0.0f;
}

__global__ __launch_bounds__(32)
void haar_level_wmma(const float* __restrict__ in, float* __restrict__ aaa,
                     DetPtrs det, int N)
{
  __shared__ float T[16 * SD];
  __shared__ float U[16 * SD];

  const int lane  = threadIdx.x;      // 0..31, full wave, EXEC all ones
  const int mrow  = lane & 15;        // M for A-frag, N for B/D-frags
  const int khalf = lane >> 4;        // K lane-half select
  const int Nh = N >> 1;
  const int tp = N >> 4;              // tiles per axis

  int t = blockIdx.x;
  const int tw = t % tp; t /= tp;
  const int th = t % tp; t /= tp;
  const int td = t;
  const int ch = blockIdx.y;          // fused (b,c): 32 channels

  const float* src = in + (size_t)ch * N * N * N
                        + ((size_t)(td * 16) * N + (size_t)(th * 16)) * N
                        + (size_t)(tw * 16);

  // ---- load 16^3 tile into LDS (float4 per lane, 32 iters) ----
  #pragma unroll 4
  for (int i = 0; i < 32; ++i) {
    int f  = lane + 32 * i;           // float4 index 0..1023
    int w4 = (f & 3) << 2;
    int h  = (f >> 2) & 15;
    int d  = f >> 6;
    float4 v = *(const float4*)(src + ((size_t)d * N + h) * N + w4);
    *(float4*)&T[d * SD + h * SH + w4] = v;
  }
  __syncthreads();

  // ---- constant B fragments = H^T chunks (identical for all 3 passes) ----
  // B frag layout (4x16, 2 VGPRs): comp j -> K = 4*kk + 2*khalf + j, N = mrow.
  v2f bf[4];
  #pragma unroll
  for (int kk = 0; kk < 4; ++kk) {
    int k0 = kk * 4 + khalf * 2;
    bf[kk].x = haarB(mrow, k0);
    bf[kk].y = haarB(mrow, k0 + 1);
  }

  // ================= pass W: T[d][h][w] -> U[d][h][w'] =================
  for (int s = 0; s < 16; ++s) {
    v8f acc = {0.f,0.f,0.f,0.f,0.f,0.f,0.f,0.f};
    #pragma unroll
    for (int kk = 0; kk < 4; ++kk) {
      int k0 = kk * 4 + khalf * 2;
      v2f a;
      a.x = T[s * SD + mrow * SH + k0];
      a.y = T[s * SD + mrow * SH + k0 + 1];
      acc = __builtin_amdgcn_wmma_f32_16x16x4_f32(false, a, false, bf[kk],
                                                  (short)0, acc, false, false);
    }
    #pragma unroll
    for (int v = 0; v < 8; ++v)                     // D: M=v+8*khalf, N=mrow
      U[s * SD + (v + 8 * khalf) * SH + mrow] = acc[v];
  }
  __syncthreads();

  // ================= pass H: U[d][h][w'] -> T[d][h'][w'] =================
  for (int s = 0; s < 16; ++s) {
    v8f acc = {0.f,0.f,0.f,0.f,0.f,0.f,0.f,0.f};
    #pragma unroll
    for (int kk = 0; kk < 4; ++kk) {
      int k0 = kk * 4 + khalf * 2;
      v2f a;                                        // A[m][k] = U[s][k][m]
      a.x = U[s * SD + k0 * SH + mrow];
      a.y = U[s * SD + (k0 + 1) * SH + mrow];
      acc = __builtin_amdgcn_wmma_f32_16x16x4_f32(false, a, false, bf[kk],
                                                  (short)0, acc, false, false);
    }
    #pragma unroll
    for (int v = 0; v < 8; ++v)                     // T[d][h'=mrow][w'=M]
      T[s * SD + mrow * SH + (v + 8 * khalf)] = acc[v];
  }
  __syncthreads();

  // ========== pass D: T[d][h'][w'] -> U[h'][d'][w'] (s = h') ==========
  for (int s = 0; s < 16; ++s) {
    v8f acc = {0.f,0.f,0.f,0.f,0.f,0.f,0.f,0.f};
    #pragma unroll
    for (int kk = 0; kk < 4; ++kk) {
      int k0 = kk * 4 + khalf * 2;
      v2f a;                                        // A[m][k] = T[k][s][m]
      a.x = T[k0 * SD + s * SH + mrow];
      a.y = T[(k0 + 1) * SD + s * SH + mrow];
      acc = __builtin_amdgcn_wmma_f32_16x16x4_f32(false, a, false, bf[kk],
                                                  (short)0, acc, false, false);
    }
    #pragma unroll
    for (int v = 0; v < 8; ++v)                     // U[h'=s][d'=mrow][w'=M]
      U[s * SD + mrow * SH + (v + 8 * khalf)] = acc[v];
  }
  __syncthreads();

  // ---- extract 8 subbands (index<8 = lo/'a', >=8 = hi/'d' per axis) ----
  const size_t chOff = (size_t)ch * Nh * Nh * Nh;
  #pragma unroll 4
  for (int i = 0; i < 32; ++i) {
    int f  = lane + 32 * i;
    int w4 = (f & 3) << 2;
    int hh = (f >> 2) & 15;
    int dd = f >> 6;
    float4 v = *(const float4*)&U[hh * SD + dd * SH + w4];
    int sb = ((dd >> 3) << 2) | ((hh >> 3) << 1) | (w4 >> 3);  // (D,H,W) bits
    float* dst = (sb == 0) ? aaa : det.p[sb - 1];
    *(float4*)&dst[chOff
        + ((size_t)(td * 8 + (dd & 7)) * Nh + (size_t)(th * 8 + (hh & 7))) * Nh
        + (size_t)(tw * 8) + (w4 & 7)] = v;
  }
}

// Level 5: 8^3 -> 4^3 per channel, scalar (16K outputs total).
__global__ void haar_final8(const float* __restrict__ in, float* __restrict__ aaa,
                            DetPtrs det)
{
  const int N = 8, Nh = 4;
  int tid = blockIdx.x * blockDim.x + threadIdx.x;
  if (tid >= 32 * Nh * Nh * Nh) return;
  int ow = tid & 3;
  int oh = (tid >> 2) & 3;
  int od = (tid >> 4) & 3;
  int ch = tid >> 6;
  const float* src = in + (size_t)ch * N * N * N;
  float x[8];
  #pragma unroll
  for (int j = 0; j < 8; ++j) {
    int dd = (j >> 2) & 1, hh = (j >> 1) & 1, ww = j & 1;
    x[j] = src[((size_t)(2 * od + dd) * N + (2 * oh + hh)) * N + (2 * ow + ww)];
  }
  const float c3 = 0.35355339059327376f;  // (1/sqrt(2))^3
  size_t o = (size_t)ch * Nh * Nh * Nh + ((size_t)od * Nh + oh) * Nh + ow;
  #pragma unroll
  for (int sb = 0; sb < 8; ++sb) {
    float acc = 0.f;
    #pragma unroll
    for (int j = 0; j < 8; ++j) {
      int neg = ((sb & j & 4) ? 1 : 0) ^ ((sb & j & 2) ? 1 : 0) ^ ((sb & j & 1) ? 1 : 0);
      acc += neg ? -x[j] : x[j];
    }
    float* dst = (sb == 0) ? aaa : det.p[sb - 1];
    dst[o] = acc * c3;
  }
}

extern "C" void kernel_launch(void* const* d_in, const int* in_sizes, int n_in,
                              void* d_out, int out_size, void* d_ws, size_t ws_size,
                              hipStream_t stream)
{
  (void)in_sizes; (void)n_in; (void)out_size; (void)ws_size;
  const float* x = (const float*)d_in[0];   // [4,8,128,128,128] fp32; d_in[1]=level=5
  float* out = (float*)d_out;

  // Workspace ping-pong for aaa bands: ws0 (<=32MB) and ws1 (<=4MB).
  float* ws0 = (float*)d_ws;
  float* ws1 = ws0 + (size_t)32 * 64 * 64 * 64;

  long long sz[6];
  for (int lv = 1; lv <= 5; ++lv) {
    long long n = 128 >> lv;
    sz[lv] = 32LL * n * n * n;
  }
  // d_out layout: Yl, then details coarsest(level5)->finest(level1), 7 keys each.
  long long detoff[6];
  detoff[5] = sz[5];
  for (int lv = 4; lv >= 1; --lv) detoff[lv] = detoff[lv + 1] + 7 * sz[lv + 1];

  auto mkdet = [&](int lv) {
    DetPtrs d;
    for (int i = 0; i < 7; ++i) d.p[i] = out + detoff[lv] + (long long)i * sz[lv];
    return d;
  };

  haar_level_wmma<<<dim3(512, 32), 32, 0, stream>>>(x,   ws0, mkdet(1), 128);
  haar_level_wmma<<<dim3(64, 32),  32, 0, stream>>>(ws0, ws1, mkdet(2), 64);
  haar_level_wmma<<<dim3(8, 32),   32, 0, stream>>>(ws1, ws0, mkdet(3), 32);
  haar_level_wmma<<<dim3(1, 32),   32, 0, stream>>>(ws0, ws1, mkdet(4), 16);
  haar_final8   <<<8, 256, 0, stream>>>(ws1, out, mkdet(5));
}